// TripletLoss_11527692222870
// MI455X (gfx1250) — compile-verified
//
#include <hip/hip_runtime.h>

// ---------------------------------------------------------------------------
// Batch-hard triplet loss, fused:  gram tiles via V_WMMA_F32_16X16X4_F32,
// half-tile (K=64) software-pipelined B loads sized to avoid VGPR spills,
// per-anchor hardest-pos/neg folded in registers; the 8192x8192 distance
// matrix is never materialized.
// ---------------------------------------------------------------------------

typedef __attribute__((ext_vector_type(2))) float v2f;
typedef __attribute__((ext_vector_type(8))) float v8f;

static constexpr int NB = 8192;   // batch
static constexpr int ND = 128;    // dim
static constexpr float MARGIN = 1.0f;
static constexpr float FMAXV = 3.402823466e38f;

// ---- kernel 0: per-row squared norms --------------------------------------
__global__ __launch_bounds__(256) void sqnorm_kernel(const float* __restrict__ E,
                                                     float* __restrict__ sqn) {
    int r = blockIdx.x * blockDim.x + threadIdx.x;
    if (r >= NB) return;
    const float4* p = (const float4*)(E + (size_t)r * ND);
    float s = 0.0f;
#pragma unroll
    for (int i = 0; i < ND / 4; ++i) {
        float4 v = p[i];
        s += v.x * v.x + v.y * v.y + v.z * v.z + v.w * v.w;
    }
    sqn[r] = s;
}

// ---- kernel 1: fused gram (WMMA f32) + batch-hard row reductions ----------
// 64 threads = 2 waves; each wave owns 16 anchor rows, sweeps all 8192 cols.
__global__ __launch_bounds__(64) void hardest_kernel(const float* __restrict__ E,
                                                     const int* __restrict__ labels,
                                                     const float* __restrict__ sqn,
                                                     float* __restrict__ hp_out,
                                                     float* __restrict__ hn_out) {
    const int lane = threadIdx.x & 31;
    const int wave = threadIdx.x >> 5;
    const int hi   = lane >> 4;    // half-wave: 0 or 1
    const int lo   = lane & 15;
    const int row_base = blockIdx.x * 32 + wave * 16;

    // Preload A panel: 16 rows x 128 K.  ISA 16x4 f32 A layout per k-step:
    // lane holds row (lane&15), elements k = 4*kk + 2*hi + {0,1}  -> float2
    v2f a[32];
    {
        const v2f* ar = (const v2f*)(E + (size_t)(row_base + lo) * ND);
#pragma unroll
        for (int kk = 0; kk < 32; ++kk) a[kk] = ar[2 * kk + hi];
    }

    // Row metadata for the 8 M-values this lane's half covers (m = i + 8*hi)
    float rsq[8];
    int   rlab[8];
#pragma unroll
    for (int i = 0; i < 8; ++i) {
        int gm = row_base + i + 8 * hi;
        rsq[i]  = sqn[gm];
        rlab[i] = labels[gm];
    }

    float hp[8], hn[8];
#pragma unroll
    for (int i = 0; i < 8; ++i) { hp[i] = -FMAXV; hn[i] = FMAXV; }

    // ---- software pipeline at half-tile (K=64) granularity ----------------
    // lane's column within a tile: n = t*16 + lo;  B[k][n] = E[n][k]
    // half q of tile t: v2f indices 32*q + 2*j + hi, j = 0..15
    constexpr int NTILES = NB / 16;
    const v2f* brow = (const v2f*)(E + (size_t)lo * ND);   // + t*(16*ND/2) per tile
    const int  tstep = (16 * ND) / 2;                      // v2f elements per tile

    v2f bbuf[2][16];                                       // 64 VGPRs total

    // prologue: half 0 of tile 0, plus tile-0 column metadata
#pragma unroll
    for (int j = 0; j < 16; ++j) bbuf[0][j] = brow[2 * j + hi];
    float cs = sqn[lo];
    int   cl = labels[lo];

    for (int t = 0; t < NTILES; ++t) {
        const v2f* bp = brow + (size_t)t * tstep;

        // issue half-1 loads, then compute half-0 (covers their latency)
#pragma unroll
        for (int j = 0; j < 16; ++j) bbuf[1][j] = bp[32 + 2 * j + hi];

        v8f acc = {};
#pragma unroll
        for (int j = 0; j < 16; ++j)
            acc = __builtin_amdgcn_wmma_f32_16x16x4_f32(
                false, a[j], false, bbuf[0][j], (short)0, acc, false, false);

        // issue next tile's half-0 loads, then compute half-1
        if (t + 1 < NTILES) {
            const v2f* bn = bp + tstep;
#pragma unroll
            for (int j = 0; j < 16; ++j) bbuf[0][j] = bn[2 * j + hi];
        }
#pragma unroll
        for (int j = 0; j < 16; ++j)
            acc = __builtin_amdgcn_wmma_f32_16x16x4_f32(
                false, a[16 + j], false, bbuf[1][j], (short)0, acc, false, false);

        // epilogue: fold this tile into the running hardest-pos/neg
        const int   gn = t * 16 + lo;
        const float csv = cs;
        const int   clv = cl;
        if (t + 1 < NTILES) {              // prefetch next tile's column meta
            cs = sqn[gn + 16];
            cl = labels[gn + 16];
        }
#pragma unroll
        for (int i = 0; i < 8; ++i) {
            float sq = rsq[i] + csv - 2.0f * acc[i];
            sq = fmaxf(sq, 0.0f);
            float d = (sq > 0.0f) ? sqrtf(sq) : 0.0f;
            int  gm   = row_base + i + 8 * hi;
            bool same = (rlab[i] == clv);
            hp[i] = (same && (gm != gn)) ? fmaxf(hp[i], d) : hp[i];
            hn[i] = (!same)              ? fminf(hn[i], d) : hn[i];
        }
    }

    // Reduce across the 16 lanes of each half-wave (xor masks 1..8 stay inside it)
#pragma unroll
    for (int i = 0; i < 8; ++i) {
#pragma unroll
        for (int m = 1; m <= 8; m <<= 1) {
            hp[i] = fmaxf(hp[i], __shfl_xor(hp[i], m, 32));
            hn[i] = fminf(hn[i], __shfl_xor(hn[i], m, 32));
        }
    }
    if (lo == 0) {
#pragma unroll
        for (int i = 0; i < 8; ++i) {
            int gm = row_base + i + 8 * hi;
            hp_out[gm] = hp[i];
            hn_out[gm] = hn[i];
        }
    }
}

// ---- kernel 2: masked mean of relu(hp - hn + margin) ----------------------
__global__ __launch_bounds__(256) void loss_kernel(const float* __restrict__ hp,
                                                   const float* __restrict__ hn,
                                                   float* __restrict__ out) {
    __shared__ float ssum[256];
    __shared__ int   scnt[256];
    float sum = 0.0f;
    int   cnt = 0;
    for (int r = threadIdx.x; r < NB; r += 256) {
        float p = hp[r], n = hn[r];
        bool valid = (p > -1e37f) && (n < 1e37f);   // sentinels => no pos / no neg
        float t = valid ? fmaxf(p - n + MARGIN, 0.0f) : 0.0f;
        sum += t;
        cnt += valid ? 1 : 0;
    }
    ssum[threadIdx.x] = sum;
    scnt[threadIdx.x] = cnt;
    __syncthreads();
    for (int s = 128; s > 0; s >>= 1) {
        if (threadIdx.x < (unsigned)s) {
            ssum[threadIdx.x] += ssum[threadIdx.x + s];
            scnt[threadIdx.x] += scnt[threadIdx.x + s];
        }
        __syncthreads();
    }
    if (threadIdx.x == 0) {
        int c = scnt[0] > 1 ? scnt[0] : 1;
        out[0] = ssum[0] / (float)c;
    }
}

// ---------------------------------------------------------------------------
extern "C" void kernel_launch(void* const* d_in, const int* in_sizes, int n_in,
                              void* d_out, int out_size, void* d_ws, size_t ws_size,
                              hipStream_t stream) {
    const float* E      = (const float*)d_in[0];   // (8192, 128) f32
    const int*   labels = (const int*)d_in[1];     // (8192,) i32
    float*       out    = (float*)d_out;           // scalar f32

    float* sqn = (float*)d_ws;      // 8192 f32
    float* hp  = sqn + NB;          // 8192 f32
    float* hn  = hp + NB;           // 8192 f32

    sqnorm_kernel<<<NB / 256, 256, 0, stream>>>(E, sqn);
    hardest_kernel<<<NB / 32, 64, 0, stream>>>(E, labels, sqn, hp, hn);
    loss_kernel<<<1, 256, 0, stream>>>(hp, hn, out);
}